// ItemLayerEmbedding_4140348473626
// MI455X (gfx1250) — compile-verified
//
#include <hip/hip_runtime.h>
#include <hip/hip_bf16.h>
#include <stdint.h>

#define B_ 1024
#define S_ 192
#define D_ 512
#define MAX_ITEMS_ 64
#define NUM_LAYERS_ 3
#define LN_EPS_ 1e-5f

typedef float v4f __attribute__((ext_vector_type(4)));
typedef int v4i __attribute__((ext_vector_type(4)));

// Address-space-typed pointer casts for the async-to-LDS builtin
#define AS1_V4I(p) ((__attribute__((address_space(1))) v4i*)(p))
#define AS3_V4I(p) ((__attribute__((address_space(3))) v4i*)(p))

// ---------------------------------------------------------------------------
// Main kernel: one wave32 per (b,s) row; persistent grid.
// (Defined first so the disasm snippet shows the async-to-LDS staging.)
// ---------------------------------------------------------------------------
__launch_bounds__(256)
__global__ void itemlayer_ln_kernel(const float* __restrict__ tok,
                                    const int* __restrict__ mask,
                                    const float* __restrict__ item_pos,
                                    const float* __restrict__ layer_emb,
                                    const float* __restrict__ temporal,
                                    const float* __restrict__ lnw,
                                    const float* __restrict__ lnb,
                                    const float* __restrict__ combined,  // may be null
                                    const int* __restrict__ startArr,    // may be null
                                    float* __restrict__ out) {
  __shared__ __align__(16) float s_layer[NUM_LAYERS_ * D_];
  __shared__ __align__(16) float s_w[D_];
  __shared__ __align__(16) float s_b[D_];

  // ---- Stage small per-row-reused tables into LDS (async-to-LDS path) ----
#if __has_builtin(__builtin_amdgcn_global_load_async_to_lds_b128)
  for (int i = threadIdx.x; i < (NUM_LAYERS_ * D_) / 4; i += blockDim.x) {
    __builtin_amdgcn_global_load_async_to_lds_b128(
        AS1_V4I(layer_emb + i * 4), AS3_V4I(&s_layer[i * 4]), 0, 0);
  }
  for (int i = threadIdx.x; i < D_ / 4; i += blockDim.x) {
    __builtin_amdgcn_global_load_async_to_lds_b128(
        AS1_V4I(lnw + i * 4), AS3_V4I(&s_w[i * 4]), 0, 0);
    __builtin_amdgcn_global_load_async_to_lds_b128(
        AS1_V4I(lnb + i * 4), AS3_V4I(&s_b[i * 4]), 0, 0);
  }
#if __has_builtin(__builtin_amdgcn_s_wait_asynccnt)
  __builtin_amdgcn_s_wait_asynccnt(0);
#else
  asm volatile("s_wait_asynccnt 0" ::: "memory");
#endif
#else
  for (int i = threadIdx.x; i < NUM_LAYERS_ * D_; i += blockDim.x)
    s_layer[i] = layer_emb[i];
  for (int i = threadIdx.x; i < D_; i += blockDim.x) {
    s_w[i] = lnw[i];
    s_b[i] = lnb[i];
  }
#endif
  __syncthreads();

  const int lane = threadIdx.x & 31;
  const int wid = threadIdx.x >> 5;
  const int wavesPerBlock = blockDim.x >> 5;
  const int totalRows = B_ * S_;
  const int nw = gridDim.x * wavesPerBlock;

  for (int row = blockIdx.x * wavesPerBlock + wid; row < totalRows; row += nw) {
    // Prefetch the next row this wave will process (streams from HBM once).
    int nrow = row + nw;
    if (nrow < totalRows)
      __builtin_prefetch(tok + (size_t)nrow * D_ + lane * 16, 0, 0);

    v4f* orow = (v4f*)(out + (size_t)row * D_);
    const int m = mask[row];  // uniform across the wave (one row per wave)
    if (m == 0) {
      v4f z = {0.f, 0.f, 0.f, 0.f};
#pragma unroll
      for (int c = 0; c < 4; ++c)
        __builtin_nontemporal_store(z, &orow[c * 32 + lane]);
      continue;
    }

    const int b = row / S_;
    const int s = row - b * S_;
    int start;
    if (startArr) {
      start = startArr[b];
    } else {
      const int* mrow = mask + b * S_;
      int ones = 0;
      for (int t = lane; t < S_; t += 32) ones += (mrow[t] != 0);
#pragma unroll
      for (int off = 16; off > 0; off >>= 1) ones += __shfl_xor(ones, off, 32);
      start = S_ - ones;
    }
    const int pos = s - start;               // >= 0 because mask[row]==1 (left-padded)
    const int item = pos / NUM_LAYERS_;      // < 64
    const int layer = pos - item * NUM_LAYERS_;

    const v4f* trow = (const v4f*)(tok + (size_t)row * D_);
    const v4f* crow;
    const v4f* crow2 = nullptr;
    if (combined) {
      crow = (const v4f*)(combined + item * D_);
    } else {
      crow = (const v4f*)(item_pos + item * D_);
      crow2 = (const v4f*)(temporal + item * D_);
    }
    const v4f* lrow = (const v4f*)(s_layer + layer * D_);

    v4f vals[4];
    float sum = 0.f, sq = 0.f;
#pragma unroll
    for (int c = 0; c < 4; ++c) {
      const int f = c * 32 + lane;                    // coalesced b128 segments
      v4f x = __builtin_nontemporal_load(&trow[f]);   // stream-once data
      v4f a = crow[f];                                // L2-hot table
      if (crow2) a += crow2[f];
      x += a + lrow[f];                               // ds_load_b128
      vals[c] = x;
      sum += x.x + x.y + x.z + x.w;
      sq += x.x * x.x + x.y * x.y + x.z * x.z + x.w * x.w;
    }
#pragma unroll
    for (int off = 16; off > 0; off >>= 1) {
      sum += __shfl_xor(sum, off, 32);
      sq += __shfl_xor(sq, off, 32);
    }
    const float mean = sum * (1.f / D_);
    const float var = sq * (1.f / D_) - mean * mean;
    const float rstd = rsqrtf(var + LN_EPS_);

#pragma unroll
    for (int c = 0; c < 4; ++c) {
      const int f = c * 32 + lane;
      v4f w = ((const v4f*)s_w)[f];
      v4f bb = ((const v4f*)s_b)[f];
      v4f y = (vals[c] - mean) * rstd * w + bb;
      __builtin_nontemporal_store(y, &orow[f]);       // write-once output
    }
  }
}

// ---------------------------------------------------------------------------
// Prep kernel 1: combined[i] = item_pos_emb[i] + temporal_decay[i]  (64x512)
// ---------------------------------------------------------------------------
__global__ void prep_combined_kernel(const float* __restrict__ item_pos,
                                     const float* __restrict__ temporal,
                                     float* __restrict__ combined) {
  int i = blockIdx.x * blockDim.x + threadIdx.x;
  if (i < MAX_ITEMS_ * D_) combined[i] = item_pos[i] + temporal[i];
}

// ---------------------------------------------------------------------------
// Prep kernel 2: start[b] = S - (#ones in mask row b)   (one wave32 per row)
// ---------------------------------------------------------------------------
__global__ void prep_start_kernel(const int* __restrict__ mask,
                                  int* __restrict__ startArr) {
  int wave = (blockIdx.x * blockDim.x + threadIdx.x) >> 5;
  int lane = threadIdx.x & 31;
  if (wave >= B_) return;
  const int* mrow = mask + wave * S_;
  int ones = 0;
  for (int s = lane; s < S_; s += 32) ones += (mrow[s] != 0);
#pragma unroll
  for (int off = 16; off > 0; off >>= 1) ones += __shfl_xor(ones, off, 32);
  if (lane == 0) startArr[wave] = S_ - ones;
}

// ---------------------------------------------------------------------------
// Host launcher
// ---------------------------------------------------------------------------
extern "C" void kernel_launch(void* const* d_in, const int* in_sizes, int n_in,
                              void* d_out, int out_size, void* d_ws, size_t ws_size,
                              hipStream_t stream) {
  const float* tok = (const float*)d_in[0];
  const int* mask = (const int*)d_in[1];
  const float* item_pos = (const float*)d_in[2];
  const float* layer_emb = (const float*)d_in[3];
  const float* temporal = (const float*)d_in[4];
  const float* lnw = (const float*)d_in[5];
  const float* lnb = (const float*)d_in[6];
  float* out = (float*)d_out;

  float* combined = nullptr;
  int* startArr = nullptr;
  const size_t combinedBytes = (size_t)MAX_ITEMS_ * D_ * sizeof(float);  // 128 KB
  const size_t startBytes = (size_t)B_ * sizeof(int);                    // 4 KB
  if (ws_size >= combinedBytes + startBytes) {
    combined = (float*)d_ws;
    startArr = (int*)((char*)d_ws + combinedBytes);
  } else if (ws_size >= startBytes) {
    startArr = (int*)d_ws;
  }

  if (combined) {
    prep_combined_kernel<<<(MAX_ITEMS_ * D_ + 255) / 256, 256, 0, stream>>>(
        item_pos, temporal, combined);
  }
  if (startArr) {
    prep_start_kernel<<<(B_ * 32 + 255) / 256, 256, 0, stream>>>(mask, startArr);
  }

  itemlayer_ln_kernel<<<2048, 256, 0, stream>>>(
      tok, mask, item_pos, layer_emb, temporal, lnw, lnb, combined, startArr, out);
}